// WienerDeblurNet_82429012345558
// MI455X (gfx1250) — compile-verified
//
#include <hip/hip_runtime.h>
#include <stdint.h>

// ---------------------------------------------------------------------------
// WienerDeblurNet on MI455X (gfx1250).
// FFTs of size 528 (=16*33) are dense DFT-matrix GEMMs -> all heavy math on
// V_WMMA_F32_16X16X4_F32.  ~0.3 TFLOP fp32 on a ~160MB working set resident
// in the 192MB L2 -> compute bound on the fp32 matrix pipe.
// GEMM blocks stage the shared B k-panel into LDS with CDNA5 async
// global->LDS copies (ASYNCcnt, double buffered); fallback is a cooperative
// load+ds_store if the builtins are absent.
// ---------------------------------------------------------------------------

namespace {
constexpr int  N     = 528;              // padded FFT size (512+16)
constexpr long N2    = (long)N * N;
constexpr int  NB    = 8;                // images
constexpr int  NC    = 3;                // channels
constexpr int  NP    = NB * NC;          // 24 planes
constexpr int  HH    = 512;
constexpr int  KHH   = 17;               // blur kernel size
constexpr int  RK    = 5;                // regularizer size
constexpr int  FR    = 24;               // regularizer count
constexpr int  DD    = 4;
constexpr int  PADP  = 8;                // (17+1)/2 - 1 (all four sides)
constexpr int  RPADP = 2;                // 5 // 2
constexpr int  MT    = N / 16;           // 33 tiles per dim
constexpr int  NCH   = N / 16;           // 33 k-chunks of 16
}

typedef __attribute__((ext_vector_type(2))) float v2f;
typedef __attribute__((ext_vector_type(8))) float v8f;
typedef __attribute__((ext_vector_type(2))) int   v2i;

// ---- CDNA5 async global->LDS staging (guarded: falls back to ds_store) ----
#if __has_builtin(__builtin_amdgcn_global_load_async_to_lds_b64) && \
    __has_builtin(__builtin_amdgcn_s_wait_asynccnt)
#define HAVE_ASYNC_LDS 1
#else
#define HAVE_ASYNC_LDS 0
#endif

#if HAVE_ASYNC_LDS
// builtin signature (from hipcc diagnostic): (v2i AS1*, v2i AS3*, imm, imm)
typedef __attribute__((address_space(1))) v2i* g_v2i_ptr;
typedef __attribute__((address_space(3))) v2i* l_v2i_ptr;
#define WAIT_ASYNC(n) __builtin_amdgcn_s_wait_asynccnt(n)
#else
#define WAIT_ASYNC(n)
#endif

// copy 8 bytes global -> LDS (async DMA when available)
__device__ __forceinline__ void copy8_to_lds(const float* g, float* l) {
#if HAVE_ASYNC_LDS
  __builtin_amdgcn_global_load_async_to_lds_b64(
      (g_v2i_ptr)(uintptr_t)g,
      (l_v2i_ptr)(unsigned)(uintptr_t)l,   // flat LDS addr[31:0] == LDS offset
      0, 0);
#else
  *(v2f*)l = *(const v2f*)g;
#endif
}

// ---------------------------------------------------------------------------
// DFT matrices: forward W[j,k] = exp(-2*pi*i*j*k/N),
// inverse Wi[j,k] = exp(+2*pi*i*j*k/N)/N  (1/N per pass -> 1/N^2 for ifft2).
// ---------------------------------------------------------------------------
__global__ void build_dft(float* wfr, float* wfi, float* wir, float* wii) {
  long idx = (long)blockIdx.x * blockDim.x + threadIdx.x;
  if (idx >= N2) return;
  int j = (int)(idx / N), k = (int)(idx % N);
  long ph = ((long)j * (long)k) % N;
  double ang = -2.0 * 3.14159265358979323846 * (double)ph / (double)N;
  float c = (float)cos(ang), s = (float)sin(ang);
  wfr[idx] = c;            wfi[idx] = s;
  wir[idx] = c / (float)N; wii[idx] = -s / (float)N;
}

// symmetric (MATLAB-style, edge-inclusive) pad 512^2 -> 528^2
__global__ void pad_symmetric(const float* __restrict__ img, float* __restrict__ xp) {
  long idx = (long)blockIdx.x * blockDim.x + threadIdx.x;
  if (idx >= (long)NP * N2) return;
  int p = (int)(idx / N2);
  int r = (int)((idx % N2) / N), c = (int)(idx % N);
  int si = r - PADP; si = (si < 0) ? (-1 - si) : (si >= HH ? 2 * HH - 1 - si : si);
  int sj = c - PADP; sj = (sj < 0) ? (-1 - sj) : (sj >= HH ? 2 * HH - 1 - sj : sj);
  xp[idx] = img[(long)p * HH * HH + (long)si * HH + sj];
}

// circular autocorrelation of PSF row/col projections over n = N-1 = 527;
// 2*17 < 527 so circular == linear, only lags 0..16 nonzero; normalized.
__global__ void autocorr_kernel(const float* __restrict__ psf, float* __restrict__ AC) {
  int b = blockIdx.x;
  if (threadIdx.x != 0) return;
  const float* p = psf + (long)b * KHH * KHH;
  float pr[KHH], pc[KHH];
  for (int i = 0; i < KHH; ++i) { float s = 0; for (int j = 0; j < KHH; ++j) s += p[i * KHH + j]; pr[i] = s; }
  for (int j = 0; j < KHH; ++j) { float s = 0; for (int i = 0; i < KHH; ++i) s += p[i * KHH + j]; pc[j] = s; }
  float* a0 = AC + (long)b * 64;
  float* a1 = a0 + 32;
  float z0 = 1.f, z1 = 1.f;
  for (int m = 0; m < KHH; ++m) {
    float s = 0; for (int i = 0; i + m < KHH; ++i) s += pr[i] * pr[i + m];
    if (m == 0) z0 = s; a0[m] = s;
  }
  for (int m = 0; m < KHH; ++m) {
    float s = 0; for (int i = 0; i + m < KHH; ++i) s += pc[i] * pc[i + m];
    if (m == 0) z1 = s; a1[m] = s;
  }
  for (int m = 0; m < KHH; ++m) { a0[m] /= z0; a1[m] /= z1; }
}

// mean-subtract + L2-normalize + scale the 24 regularizer filters
__global__ void prep_filters(const float* __restrict__ g, const float* __restrict__ scale,
                             float* __restrict__ fw) {
  int f = blockIdx.x * blockDim.x + threadIdx.x;
  if (f >= FR) return;
  const float* gf = g + (long)f * RK * RK;
  float mean = 0;
  for (int i = 0; i < RK * RK; ++i) mean += gf[i];
  mean /= (float)(RK * RK);
  float ss = 0;
  for (int i = 0; i < RK * RK; ++i) { float w = gf[i] - mean; ss += w * w; }
  float k = scale[f] * rsqrtf(ss);
  for (int i = 0; i < RK * RK; ++i) fw[f * RK * RK + i] = (gf[i] - mean) * k;
}

// zero-pad + circshift PSF so its center lands at (0,0); imag = 0
__global__ void fill_psf(const float* __restrict__ psf, float* __restrict__ tr, float* __restrict__ ti) {
  long idx = (long)blockIdx.x * blockDim.x + threadIdx.x;
  if (idx >= (long)NB * N2) return;
  int b = (int)(idx / N2);
  int r = (int)((idx % N2) / N), c = (int)(idx % N);
  int iu = (r + PADP) % N, iv = (c + PADP) % N;
  float v = (iu < KHH && iv < KHH) ? psf[(long)b * KHH * KHH + iu * KHH + iv] : 0.f;
  tr[idx] = v; ti[idx] = 0.f;
}

__global__ void fill_g(const float* __restrict__ fw, float* __restrict__ tr, float* __restrict__ ti) {
  long idx = (long)blockIdx.x * blockDim.x + threadIdx.x;
  if (idx >= (long)FR * N2) return;
  int f = (int)(idx / N2);
  int r = (int)((idx % N2) / N), c = (int)(idx % N);
  int iu = (r + RPADP) % N, iv = (c + RPADP) % N;
  float v = (iu < RK && iv < RK) ? fw[f * RK * RK + iu * RK + iv] : 0.f;
  tr[idx] = v; ti[idx] = 0.f;
}

__global__ void greg_accum(const float* __restrict__ gr, const float* __restrict__ gi,
                           float* __restrict__ greg) {
  long idx = (long)blockIdx.x * blockDim.x + threadIdx.x;
  if (idx >= N2) return;
  float s = 0;
  for (int f = 0; f < FR; ++f) {
    float a = gr[(long)f * N2 + idx], b = gi[(long)f * N2 + idx];
    s += a * a + b * b;
  }
  greg[idx] = s;
}

__global__ void real2complex(const float* __restrict__ x, float* __restrict__ tr, float* __restrict__ ti) {
  long idx = (long)blockIdx.x * blockDim.x + threadIdx.x;
  if (idx >= (long)NP * N2) return;
  tr[idx] = x[idx]; ti[idx] = 0.f;
}

// plane-wise complex multiply by per-image OTF K
__global__ void cmul_otf(float* __restrict__ tr, float* __restrict__ ti,
                         const float* __restrict__ kr, const float* __restrict__ ki) {
  long idx = (long)blockIdx.x * blockDim.x + threadIdx.x;
  if (idx >= (long)NP * N2) return;
  int  b = (int)(idx / N2) / NC;
  long o = idx % N2;
  float xr = tr[idx], xi = ti[idx];
  float Kr = kr[(long)b * N2 + o], Ki = ki[(long)b * N2 + o];
  tr[idx] = xr * Kr - xi * Ki;
  ti[idx] = xr * Ki + xi * Kr;
}

// x = alpha*x + (1-alpha)*blurred; alpha from autocorr betas
__global__ void blend_edgetaper(float* __restrict__ xp, const float* __restrict__ blurR,
                                const float* __restrict__ AC) {
  long idx = (long)blockIdx.x * blockDim.x + threadIdx.x;
  if (idx >= (long)NP * N2) return;
  int b = (int)(idx / N2) / NC;
  int r = (int)((idx % N2) / N), c = (int)(idx % N);
  const float* a0 = AC + (long)b * 64;
  const float* a1 = a0 + 32;
  int m0 = min(r, (N - 1) - r);             // index N-1 -> lag 0 (appended z[0])
  int m1 = min(c, (N - 1) - c);
  float b0 = (m0 < KHH) ? a0[m0] : 0.f;
  float b1 = (m1 < KHH) ? a1[m1] : 0.f;
  float alpha = (1.f - b0) * (1.f - b1);
  xp[idx] = alpha * xp[idx] + (1.f - alpha) * blurR[idx];
}

// X = conj(K)*Y * sum_d w_d / (|K|^2 + exp(log_alpha_d)*Greg)
// (ifft2 commutes with the weighted d-sum -> single inverse transform)
__global__ void wiener_combine(float* __restrict__ tr, float* __restrict__ ti,
                               const float* __restrict__ kr, const float* __restrict__ ki,
                               const float* __restrict__ greg,
                               const float* __restrict__ log_alpha,
                               const float* __restrict__ wts) {
  long idx = (long)blockIdx.x * blockDim.x + threadIdx.x;
  if (idx >= (long)NP * N2) return;
  int  b = (int)(idx / N2) / NC;
  long o = idx % N2;
  float Kr = kr[(long)b * N2 + o], Ki = ki[(long)b * N2 + o];
  float k2 = Kr * Kr + Ki * Ki;
  float gg = greg[o];
  float s = 0;
  #pragma unroll
  for (int d = 0; d < DD; ++d) s += wts[d] / (k2 + __expf(log_alpha[d]) * gg);
  float yr = tr[idx], yi = ti[idx];
  tr[idx] = (Kr * yr + Ki * yi) * s;
  ti[idx] = (Kr * yi - Ki * yr) * s;
}

__global__ void crop_out(const float* __restrict__ tr, float* __restrict__ out) {
  long idx = (long)blockIdx.x * blockDim.x + threadIdx.x;
  if (idx >= (long)NP * HH * HH) return;
  int p = (int)(idx / ((long)HH * HH));
  long rem = idx % ((long)HH * HH);
  int r = (int)(rem / HH), c = (int)(rem % HH);
  out[idx] = tr[(long)p * N2 + (long)(r + PADP) * N + (c + PADP)];
}

// ---------------------------------------------------------------------------
// Batched complex GEMM  Y[z] = A[z] @ B[z]   (stride 0 => operand shared)
// All matrices N x N row-major, split real/imag planes.
//
// Block = 4 waves sharing one N-tile column of B: the 16-wide B k-chunk
// (real+imag, 2KB) is staged into LDS once per block via async global->LDS
// DMA (double buffered; completion via s_wait_asynccnt + barrier), then each
// wave feeds its V_WMMA_F32_16X16X4_F32 quartet (RR/II/RI/IR) from LDS.
// A fragments stay as contiguous per-lane b64 global loads (rows stream
// through L0/L2) with near-scope prefetch of the next panel.
// Fragment layouts per CDNA5 ISA 7.12.2 (fp32 A 16x4: lanes 0-15 K={k,k+1},
// lanes 16-31 K={k+2,k+3}; B mirrored; C/D: vgpr r, lane half h -> M=r+8h).
// ---------------------------------------------------------------------------
__global__ __launch_bounds__(128)
void cgemm(const float* __restrict__ Ar, const float* __restrict__ Ai, long aStride,
           const float* __restrict__ Br, const float* __restrict__ Bi, long bStride,
           float* __restrict__ Yr, float* __restrict__ Yi, long yStride) {
  __shared__ float lbr[2][16][16];
  __shared__ float lbi[2][16][16];

  int mtile = blockIdx.y * 4 + threadIdx.y;            // wave-uniform
  int mtc   = (mtile < MT) ? mtile : (MT - 1);         // clamp: all waves cooperate
  int ntile = blockIdx.x;
  int bat   = blockIdx.z;

  const float* ar = Ar + (long)bat * aStride;
  const float* ai = Ai + (long)bat * aStride;
  const float* br = Br + (long)bat * bStride;
  const float* bi = Bi + (long)bat * bStride;
  float* yr = Yr + (long)bat * yStride;
  float* yi = Yi + (long)bat * yStride;

  int lane = threadIdx.x;
  int half = lane >> 4;
  int l15  = lane & 15;
  int m    = mtc * 16 + l15;
  int n0   = ntile * 16;

  // cooperative B staging role: 128 threads x 8B cover one 16x16 fp32 chunk
  int t    = threadIdx.y * 32 + lane;
  int srow = t >> 3;
  int scol = (t & 7) * 2;

  const float* arow_r = ar + (long)m * N;
  const float* arow_i = ai + (long)m * N;

  v8f RR = {}, II = {}, RI = {}, IR = {};

  // prologue: stage chunk 0 into buffer 0
  copy8_to_lds(br + (long)srow * N + n0 + scol, &lbr[0][srow][scol]);
  copy8_to_lds(bi + (long)srow * N + n0 + scol, &lbi[0][srow][scol]);

  for (int kc = 0; kc < NCH; ++kc) {
    int cur = kc & 1;
    if (kc + 1 < NCH) {                                 // stage next chunk
      long krow = (long)((kc + 1) * 16 + srow) * N;
      copy8_to_lds(br + krow + n0 + scol, &lbr[cur ^ 1][srow][scol]);
      copy8_to_lds(bi + krow + n0 + scol, &lbi[cur ^ 1][srow][scol]);
      WAIT_ASYNC(2);                                    // chunk kc landed, kc+1 in flight
    } else {
      WAIT_ASYNC(0);
    }
    __syncthreads();

    int k0 = kc * 16;
    #pragma unroll
    for (int ks = 0; ks < 16; ks += 4) {
      int ka = ks + 2 * half;
      v2f fa_r = *(const v2f*)(arow_r + k0 + ka);       // contiguous b64 loads
      v2f fa_i = *(const v2f*)(arow_i + k0 + ka);
      v2f fb_r, fb_i;
      fb_r.x = lbr[cur][ka][l15];     fb_r.y = lbr[cur][ka + 1][l15];
      fb_i.x = lbi[cur][ka][l15];     fb_i.y = lbi[cur][ka + 1][l15];
      if (ks == 0 && k0 + 32 < N) {                     // near-scope prefetch of A
        __builtin_prefetch(arow_r + k0 + 32, 0, 3);
        __builtin_prefetch(arow_i + k0 + 32, 0, 3);
      }
      RR = __builtin_amdgcn_wmma_f32_16x16x4_f32(false, fa_r, false, fb_r, (short)0, RR, false, false);
      II = __builtin_amdgcn_wmma_f32_16x16x4_f32(false, fa_i, false, fb_i, (short)0, II, false, false);
      RI = __builtin_amdgcn_wmma_f32_16x16x4_f32(false, fa_r, false, fb_i, (short)0, RI, false, false);
      IR = __builtin_amdgcn_wmma_f32_16x16x4_f32(false, fa_i, false, fb_r, (short)0, IR, false, false);
    }
    __syncthreads();                                    // before buf[cur] is re-staged
  }

  if (mtile < MT) {                                     // wave-uniform store guard
    v8f outR = RR - II;
    v8f outI = RI + IR;
    #pragma unroll
    for (int r = 0; r < 8; ++r) {
      long row = (long)(mtile * 16 + r + 8 * half);
      yr[row * N + n0 + l15] = outR[r];
      yi[row * N + n0 + l15] = outI[r];
    }
  }
}

// ---------------------------------------------------------------------------
extern "C" void kernel_launch(void* const* d_in, const int* in_sizes, int n_in,
                              void* d_out, int out_size, void* d_ws, size_t ws_size,
                              hipStream_t stream) {
  (void)in_sizes; (void)n_in; (void)out_size; (void)ws_size;
  const float* image    = (const float*)d_in[0];
  const float* blurK    = (const float*)d_in[1];
  /* d_in[2] = stdn : unused by the reference math */
  const float* gweights = (const float*)d_in[3];
  const float* gscale   = (const float*)d_in[4];
  const float* logalpha = (const float*)d_in[5];
  const float* mixw     = (const float*)d_in[6];
  float* out = (float*)d_out;

  float* ws   = (float*)d_ws;
  float* WFR  = ws;             float* WFI  = WFR  + N2;
  float* WIR_ = WFI + N2;       float* WII_ = WIR_ + N2;
  float* XP   = WII_ + N2;                              // NP planes (real)
  float* KR   = XP  + (long)NP * N2;                    // NB planes
  float* KI   = KR  + (long)NB * N2;
  float* GREG = KI  + (long)NB * N2;                    // 1 plane
  float* T1R  = GREG + N2;      float* T1I = T1R + (long)NP * N2;
  float* T2R  = T1I + (long)NP * N2;
  float* T2I  = T2R + (long)NP * N2;
  float* AC   = T2I + (long)NP * N2;                    // NB*64
  float* FWB  = AC  + (long)NB * 64;                    // FR*25

  dim3 eb(256);
  auto g1 = [](long n) { return dim3((unsigned)((n + 255) / 256)); };
  dim3 gmm(MT, (MT + 3) / 4, 1);
  dim3 bmm(32, 4);

  // setup
  build_dft<<<g1(N2), eb, 0, stream>>>(WFR, WFI, WIR_, WII_);
  pad_symmetric<<<g1((long)NP * N2), eb, 0, stream>>>(image, XP);
  autocorr_kernel<<<dim3(NB), dim3(32), 0, stream>>>(blurK, AC);
  prep_filters<<<dim3(1), dim3(32), 0, stream>>>(gweights, gscale, FWB);

  // K = fft2(pad+roll(psf))  -- batch NB
  fill_psf<<<g1((long)NB * N2), eb, 0, stream>>>(blurK, T1R, T1I);
  { dim3 g = gmm; g.z = NB;
    cgemm<<<g, bmm, 0, stream>>>(WFR, WFI, 0L, T1R, T1I, N2, T2R, T2I, N2);
    cgemm<<<g, bmm, 0, stream>>>(T2R, T2I, N2, WFR, WFI, 0L, KR, KI, N2);
  }

  // Greg = sum_f |fft2(reg filter)|^2  (image independent) -- batch FR
  fill_g<<<g1((long)FR * N2), eb, 0, stream>>>(FWB, T1R, T1I);
  { dim3 g = gmm; g.z = FR;
    cgemm<<<g, bmm, 0, stream>>>(WFR, WFI, 0L, T1R, T1I, N2, T2R, T2I, N2);
    cgemm<<<g, bmm, 0, stream>>>(T2R, T2I, N2, WFR, WFI, 0L, T1R, T1I, N2);
  }
  greg_accum<<<g1(N2), eb, 0, stream>>>(T1R, T1I, GREG);

  // edgetaper: blurred = ifft2(fft2(xp) * K), then alpha-blend
  real2complex<<<g1((long)NP * N2), eb, 0, stream>>>(XP, T1R, T1I);
  { dim3 g = gmm; g.z = NP;
    cgemm<<<g, bmm, 0, stream>>>(WFR, WFI, 0L, T1R, T1I, N2, T2R, T2I, N2);
    cgemm<<<g, bmm, 0, stream>>>(T2R, T2I, N2, WFR, WFI, 0L, T1R, T1I, N2);
  }
  cmul_otf<<<g1((long)NP * N2), eb, 0, stream>>>(T1R, T1I, KR, KI);
  { dim3 g = gmm; g.z = NP;
    cgemm<<<g, bmm, 0, stream>>>(WIR_, WII_, 0L, T1R, T1I, N2, T2R, T2I, N2);
    cgemm<<<g, bmm, 0, stream>>>(T2R, T2I, N2, WIR_, WII_, 0L, T1R, T1I, N2);
  }
  blend_edgetaper<<<g1((long)NP * N2), eb, 0, stream>>>(XP, T1R, AC);

  // Wiener: Y = fft2(x); Xsum = conj(K)*Y * sum_d w_d/S_d ; out = ifft2(Xsum)
  real2complex<<<g1((long)NP * N2), eb, 0, stream>>>(XP, T1R, T1I);
  { dim3 g = gmm; g.z = NP;
    cgemm<<<g, bmm, 0, stream>>>(WFR, WFI, 0L, T1R, T1I, N2, T2R, T2I, N2);
    cgemm<<<g, bmm, 0, stream>>>(T2R, T2I, N2, WFR, WFI, 0L, T1R, T1I, N2);
  }
  wiener_combine<<<g1((long)NP * N2), eb, 0, stream>>>(T1R, T1I, KR, KI, GREG, logalpha, mixw);
  { dim3 g = gmm; g.z = NP;
    cgemm<<<g, bmm, 0, stream>>>(WIR_, WII_, 0L, T1R, T1I, N2, T2R, T2I, N2);
    cgemm<<<g, bmm, 0, stream>>>(T2R, T2I, N2, WIR_, WII_, 0L, T1R, T1I, N2);
  }
  crop_out<<<g1((long)NP * HH * HH), eb, 0, stream>>>(T1R, out);
}